// PennylaneQuanvolutionFilter_65481071409719
// MI455X (gfx1250) — compile-verified
//
#include <hip/hip_runtime.h>
#include <hip/hip_bf16.h>

typedef float v2f __attribute__((ext_vector_type(2)));
typedef float v8f __attribute__((ext_vector_type(8)));

#define NQ 4
#define NL 3

// ---------------------------------------------------------------------------
// Kernel 1 (tiny, one block): build Astack[k][n] (16 x 64 f32) in d_ws.
//   A_q[a][b] = Re( sum_s conj(U[s,a]) * z_q(s) * U[s,b] )
//   Astack[a][q*16+b] = A_q[a][b]
// Threads 0..15 each simulate one basis column of U through the 3 layers.
// ---------------------------------------------------------------------------
__global__ void __launch_bounds__(256) quanv_prep_kernel(
    const float* __restrict__ w, float* __restrict__ Astack) {
  __shared__ float Ure[16][16];
  __shared__ float Uim[16][16];
  const int tid = threadIdx.x;

  if (tid < 16) {
    float re[16], im[16];
    #pragma unroll
    for (int s = 0; s < 16; ++s) { re[s] = (s == tid) ? 1.f : 0.f; im[s] = 0.f; }

    for (int layer = 0; layer < NL; ++layer) {
      for (int q = 0; q < NQ; ++q) {
        const float phi = w[layer * 12 + q * 3 + 0];
        const float th  = w[layer * 12 + q * 3 + 1];
        const float om  = w[layer * 12 + q * 3 + 2];
        float ct, st;            // cos/sin(theta/2)
        __sincosf(0.5f * th, &st, &ct);
        float cpp, spp, cpm, spm;
        __sincosf(0.5f * (phi + om), &spp, &cpp);   // e_pp = cpp + i*spp
        __sincosf(0.5f * (phi - om), &spm, &cpm);   // e_pm = cpm + i*spm
        // Rot matrix: [[e_mm*c, -e_pm*s], [e_mp*s, e_pp*c]]
        const float m00r =  cpp * ct, m00i = -spp * ct;   // e_mm*c
        const float m01r = -cpm * st, m01i = -spm * st;   // -e_pm*s
        const float m10r =  cpm * st, m10i = -spm * st;   // e_mp*s
        const float m11r =  cpp * ct, m11i =  spp * ct;   // e_pp*c
        const int mask = 8 >> q;    // wire q <-> bit (3-q)
        #pragma unroll
        for (int lo = 0; lo < 16; ++lo) {
          if (lo & mask) continue;
          const int hi = lo | mask;
          const float ar = re[lo], ai = im[lo], br = re[hi], bi = im[hi];
          re[lo] = m00r * ar - m00i * ai + m01r * br - m01i * bi;
          im[lo] = m00r * ai + m00i * ar + m01r * bi + m01i * br;
          re[hi] = m10r * ar - m10i * ai + m11r * br - m11i * bi;
          im[hi] = m10r * ai + m10i * ar + m11r * bi + m11i * br;
        }
      }
      // CNOT ring: (0,1),(1,2),(2,3),(3,0)
      const int ctrl[4] = {0, 1, 2, 3}, tgt[4] = {1, 2, 3, 0};
      #pragma unroll
      for (int e = 0; e < 4; ++e) {
        const int cm = 8 >> ctrl[e], tm = 8 >> tgt[e];
        #pragma unroll
        for (int s = 0; s < 16; ++s) {
          if ((s & cm) && !(s & tm)) {
            const int s2 = s | tm;
            float t = re[s]; re[s] = re[s2]; re[s2] = t;
            t = im[s]; im[s] = im[s2]; im[s2] = t;
          }
        }
      }
    }
    #pragma unroll
    for (int s = 0; s < 16; ++s) { Ure[s][tid] = re[s]; Uim[s][tid] = im[s]; }
  }
  __syncthreads();

  // 1024 entries of Astack over 256 threads
  for (int e = tid; e < 1024; e += 256) {
    const int a = e >> 6;           // 0..15  (k / row)
    const int n = e & 63;           // 0..63  (q*16 + b)
    const int q = n >> 4, b = n & 15;
    const int qm = 8 >> q;
    float acc = 0.f;
    #pragma unroll
    for (int s = 0; s < 16; ++s) {
      const float z = (s & qm) ? -1.f : 1.f;
      acc += z * (Ure[s][a] * Ure[s][b] + Uim[s][a] * Uim[s][b]);
    }
    Astack[e] = acc;
  }
}

// ---------------------------------------------------------------------------
// Kernel 2: main pass. One wave = one 16-patch tile.
//   T (16x64) = S0 (16x16) @ Astack (16x64) via 16x V_WMMA_F32_16X16X4_F32
//   exp_q[p]  = sum_n T[p][q*16+n] * s0[p][n]  (via LDS transpose staging)
// ---------------------------------------------------------------------------
__global__ void __launch_bounds__(256) quanv_main_kernel(
    const float* __restrict__ x, const float* __restrict__ Astack,
    float* __restrict__ out) {
  __shared__ float ldsT[8][4][16][16];   // per-wave 4KB -> 32KB per block

  const int tid  = threadIdx.x;
  const int wave = tid >> 5;
  const int lane = tid & 31;
  const int half = lane >> 4;            // which lane half (K split)
  const int n    = lane & 15;            // row-of-S0 / col-of-B index

  const long tile  = (long)blockIdx.x * 8 + wave;
  const long patch = tile * 16 + n;      // this lane's patch
  const int  bimg  = (int)(patch / 196);
  const int  p     = (int)(patch % 196);
  const int  pi    = p / 14, pj = p % 14;

  // 2x2 patch pixels: [2i,2j],[2i,2j+1],[2i+1,2j],[2i+1,2j+1]
  const float* img = x + (long)bimg * 784 + pi * 56 + pj * 2;
  const float f0 = img[0], f1 = img[1], f2 = img[28], f3 = img[29];

  // half-angles: theta_q/2 = pi/2 * f_q
  const float HPI = 1.57079632679489662f;
  float c0, s0v, c1, s1v, c2, s2v, c3, s3v;
  __sincosf(f0 * HPI, &s0v, &c0);
  __sincosf(f1 * HPI, &s1v, &c1);
  __sincosf(f2 * HPI, &s2v, &c2);
  __sincosf(f3 * HPI, &s3v, &c3);

  // s0[idx] = pHi[idx>>2] * pLo[idx&3]   (idx bit3 = qubit0 ... bit0 = qubit3)
  const float pHi[4] = { c0 * c1, c0 * s1v, s0v * c1, s0v * s1v };
  const float pLo[4] = { c2 * c3, c2 * s3v, s2v * c3, s2v * s3v };

  const int kc = 2 * half;               // K offset for this lane half

  v8f acc0 = {}, acc1 = {}, acc2 = {}, acc3 = {};
  #pragma unroll
  for (int j = 0; j < 4; ++j) {
    // A fragment: VGPR0 = s0[4j+kc], VGPR1 = s0[4j+kc+1], row M = n
    v2f av;
    av.x = pHi[j] * pLo[kc];
    av.y = pHi[j] * pLo[kc + 1];
    const int k0 = 4 * j + kc;
    // B fragments: VGPR0 = Astack[k0][col], VGPR1 = Astack[k0+1][col], col N = n
    v2f b0, b1, b2, b3;
    b0.x = Astack[k0 * 64 +      n];  b0.y = Astack[(k0 + 1) * 64 +      n];
    b1.x = Astack[k0 * 64 + 16 + n];  b1.y = Astack[(k0 + 1) * 64 + 16 + n];
    b2.x = Astack[k0 * 64 + 32 + n];  b2.y = Astack[(k0 + 1) * 64 + 32 + n];
    b3.x = Astack[k0 * 64 + 48 + n];  b3.y = Astack[(k0 + 1) * 64 + 48 + n];
    acc0 = __builtin_amdgcn_wmma_f32_16x16x4_f32(false, av, false, b0, (short)0, acc0, false, false);
    acc1 = __builtin_amdgcn_wmma_f32_16x16x4_f32(false, av, false, b1, (short)0, acc1, false, false);
    acc2 = __builtin_amdgcn_wmma_f32_16x16x4_f32(false, av, false, b2, (short)0, acc2, false, false);
    acc3 = __builtin_amdgcn_wmma_f32_16x16x4_f32(false, av, false, b3, (short)0, acc3, false, false);
  }

  // D layout: VGPR v holds row (v + 8*half), column n -> stage to LDS
  #pragma unroll
  for (int v = 0; v < 8; ++v) {
    const int row = v + 8 * half;
    ldsT[wave][0][row][n] = acc0[v];
    ldsT[wave][1][row][n] = acc1[v];
    ldsT[wave][2][row][n] = acc2[v];
    ldsT[wave][3][row][n] = acc3[v];
  }
  __syncthreads();

  // Row dots: lane handles its patch (row n), qubits {2*half, 2*half+1}
  float sum0 = 0.f, sum1 = 0.f;
  #pragma unroll
  for (int k = 0; k < 16; ++k) {
    const float sv = pHi[k >> 2] * pLo[k & 3];
    sum0 += ldsT[wave][2 * half    ][n][k] * sv;
    sum1 += ldsT[wave][2 * half + 1][n][k] * sv;
  }
  // out flat index = patch*4 + q ; q in {2*half, 2*half+1} -> packed float2
  *(float2*)(out + patch * 4 + 2 * half) = make_float2(sum0, sum1);
}

extern "C" void kernel_launch(void* const* d_in, const int* in_sizes, int n_in,
                              void* d_out, int out_size, void* d_ws, size_t ws_size,
                              hipStream_t stream) {
  const float* x = (const float*)d_in[0];     // (4096,1,28,28) f32
  const float* w = (const float*)d_in[1];     // (3,4,3) f32
  float* out     = (float*)d_out;             // (4096,784) f32
  float* Astack  = (float*)d_ws;              // 16*64 f32 = 4KB scratch

  // total patches = pixels/4 ; one wave per 16 patches ; 8 waves per block
  const long n_pixels  = in_sizes[0];
  const long n_patches = n_pixels / 4;        // 802816
  const int  n_blocks  = (int)(n_patches / 128); // 6272 (exact)

  hipLaunchKernelGGL(quanv_prep_kernel, dim3(1), dim3(256), 0, stream, w, Astack);
  hipLaunchKernelGGL(quanv_main_kernel, dim3(n_blocks), dim3(256), 0, stream,
                     x, Astack, out);
}